// GraphWaveNet_47055661695436
// MI455X (gfx1250) — compile-verified
//
#include <hip/hip_runtime.h>

typedef float v2f __attribute__((ext_vector_type(2)));
typedef float v8f __attribute__((ext_vector_type(8)));

static constexpr int NN   = 20000;
static constexpr int TT0  = 13;
static constexpr int EE   = 320000;
static constexpr int LLAY = 8;

// ---------------------------------------------------------------------------
// Templated f32 WMMA GEMM: Cout[M,Nc] = act( A[M,K]*B[K,Nc] + bias + Cin )
//   A addressed as A[row*lda + aoff + k]           (lda, aoff even => 8B aligned)
//   BT==1  -> B stored [Nc,K] row-major (transposed weights), v2f loads
//   NPAD=1 -> Nc not multiple of 16: mask B columns (branch-free), guard stores
//   LDSB=1 -> stage all of B (K*Nc <= 8192 floats) into LDS once per block
//   KT     -> compile-time K: fully unrolled chain of v_wmma_f32_16x16x4_f32
// ---------------------------------------------------------------------------
template <int BT, int NPAD, int LDSB, int KT>
__global__ __launch_bounds__(256) void k_gemm_t(
    const float* __restrict__ A, int lda, int aoff,
    const float* __restrict__ B,
    const float* __restrict__ bias,
    const float* Cin, float* Cout, int ldc,
    int M, int Nc, int act)
{
  __shared__ float smem[LDSB ? 8192 : 1];
  if (LDSB) {
    int total = KT * Nc;
    for (int i = threadIdx.x; i < total; i += 256) smem[i] = B[i];
    __syncthreads();
  }

  int wave = (int)((blockIdx.x * blockDim.x + threadIdx.x) >> 5);
  int lane = threadIdx.x & 31;
  int ntiles = (Nc + 15) >> 4;
  int mt = wave / ntiles;
  int nt = wave - mt * ntiles;
  if (mt * 16 >= M) return;
  int m0 = mt * 16, n0 = nt * 16;
  int lm  = lane & 15;
  int lk2 = (lane >> 4) * 2;            // 0 for lanes 0-15, 2 for lanes 16-31

  int row_a = m0 + lm;                  // A fragment row
  int col_b = n0 + lm;                  // B fragment column
  int cb    = NPAD ? min(col_b, Nc - 1) : col_b;   // clamp: no OOB reads
  float bmask = (!NPAD || col_b < Nc) ? 1.f : 0.f;

  const float* arow = A + (size_t)row_a * lda + aoff;
  const float* BbT  = BT ? (LDSB ? (smem + (size_t)cb * KT) : (B + (size_t)cb * KT))
                         : nullptr;

  v8f c = {0.f,0.f,0.f,0.f,0.f,0.f,0.f,0.f};
#pragma unroll
  for (int k0 = 0; k0 < KT; k0 += 4) {
    int ka = k0 + lk2;
    v2f a = *(const v2f*)(arow + ka);
    v2f b;
    if (BT) {
      b = *(const v2f*)(BbT + ka);
      if (NPAD) { b.x *= bmask; b.y *= bmask; }
    } else {
      if (LDSB) { b.x = smem[ka * Nc + cb];         b.y = smem[(ka + 1) * Nc + cb]; }
      else      { b.x = B[(size_t)ka * Nc + cb];    b.y = B[(size_t)(ka + 1) * Nc + cb]; }
    }
    c = __builtin_amdgcn_wmma_f32_16x16x4_f32(false, a, false, b, (short)0, c, false, false);
  }

  // Epilogue: lane&15 = column; vgpr j = row (lanes>=16 own rows m0+8..m0+15)
  int colw  = n0 + (lane & 15);
  int rbase = m0 + ((lane >> 4) << 3);
  if (!NPAD || colw < Nc) {
    float bv = bias ? bias[colw] : 0.f;
#pragma unroll
    for (int j = 0; j < 8; ++j) {
      int row = rbase + j;
      float v = c[j] + bv;
      if (Cin) v += Cin[(size_t)row * ldc + colw];
      if (act == 1) v = fmaxf(v, 0.f);
      Cout[(size_t)row * ldc + colw] = v;
    }
  }
}

// ---------------------------------------------------------------------------
// Fused dilated gated conv: h[(n,tt),c] = tanh(Xcat*Wf + fb) * sigmoid(Xcat*Wg + gb)
//   Xcat[(n,tt), k*32+i] = x[n, tt+k*d, i]   (x layout [N, Tin, 32])
//   Weights pre-transposed into LDS as W[kk=k*32+i][c]; K=64 fully unrolled.
// ---------------------------------------------------------------------------
__global__ __launch_bounds__(256) void k_gatedconv(
    const float* __restrict__ x, int Tin, int d,
    const float* __restrict__ fw, const float* __restrict__ fb,
    const float* __restrict__ gw, const float* __restrict__ gb,
    float* __restrict__ h, int tout, int M)
{
  __shared__ float sWf[64 * 32];
  __shared__ float sWg[64 * 32];
  for (int i = threadIdx.x; i < 2048; i += 256) {
    int kk = i >> 5, cc = i & 31;
    int ii = kk & 31, kc = kk >> 5;
    sWf[i] = fw[(size_t)(cc * 32 + ii) * 2 + kc];
    sWg[i] = gw[(size_t)(cc * 32 + ii) * 2 + kc];
  }
  __syncthreads();

  int wave = (int)((blockIdx.x * blockDim.x + threadIdx.x) >> 5);
  int lane = threadIdx.x & 31;
  int mt = wave >> 1;
  int nt = wave & 1;
  if (mt * 16 >= M) return;
  int m0 = mt * 16, n0 = nt * 16;
  int lm  = lane & 15;
  int lk2 = (lane >> 4) * 2;

  int row = m0 + lm;
  int n  = row / tout;
  int tt = row - n * tout;
  const float* xrow = x + (size_t)n * Tin * 32;
  int col = n0 + lm;

  v8f cf = {0.f,0.f,0.f,0.f,0.f,0.f,0.f,0.f};
  v8f cg = {0.f,0.f,0.f,0.f,0.f,0.f,0.f,0.f};
#pragma unroll
  for (int k0 = 0; k0 < 64; k0 += 4) {
    int kk = k0 + lk2;                 // lane's K pair (same conv tap: ii even, ii+1)
    int kc = kk >> 5;
    int ii = kk & 31;
    v2f a = *(const v2f*)(xrow + (size_t)(tt + kc * d) * 32 + ii);
    v2f bf, bg;
    bf.x = sWf[kk * 32 + col];  bf.y = sWf[(kk + 1) * 32 + col];
    bg.x = sWg[kk * 32 + col];  bg.y = sWg[(kk + 1) * 32 + col];
    cf = __builtin_amdgcn_wmma_f32_16x16x4_f32(false, a, false, bf, (short)0, cf, false, false);
    cg = __builtin_amdgcn_wmma_f32_16x16x4_f32(false, a, false, bg, (short)0, cg, false, false);
  }

  int colw  = n0 + (lane & 15);
  int rbase = m0 + ((lane >> 4) << 3);
  float fbv = fb[colw], gbv = gb[colw];
#pragma unroll
  for (int j = 0; j < 8; ++j) {
    int r = rbase + j;
    float f = tanhf(cf[j] + fbv);
    float z = cg[j] + gbv;
    float g = 1.f / (1.f + __expf(-z));
    h[(size_t)r * 32 + colw] = f * g;
  }
}

// ---------------------------------------------------------------------------
// Elementwise / reduction / scatter kernels
// ---------------------------------------------------------------------------
__global__ __launch_bounds__(256) void k_zero4(float4* __restrict__ p, long long n4) {
  long long i = (long long)blockIdx.x * blockDim.x + threadIdx.x;
  if (i < n4) p[i] = make_float4(0.f, 0.f, 0.f, 0.f);
}

__global__ __launch_bounds__(256) void k_degcount(const int* __restrict__ dst,
                                                  float* __restrict__ deg, int e) {
  int i = blockIdx.x * blockDim.x + threadIdx.x;
  if (i < e) atomicAdd(&deg[dst[i]], 1.f);
}

__global__ __launch_bounds__(256) void k_deginv(float* __restrict__ deg, int n) {
  int i = blockIdx.x * blockDim.x + threadIdx.x;
  if (i < n) deg[i] = 1.f / fmaxf(deg[i], 1.f);
}

__global__ __launch_bounds__(256) void k_enter(const float* __restrict__ in,
                                               const float* __restrict__ w,
                                               const float* __restrict__ b,
                                               float* __restrict__ x, long long total) {
  long long i = (long long)blockIdx.x * blockDim.x + threadIdx.x;
  if (i >= total) return;
  int c = (int)(i & 31);
  long long r = i >> 5;                       // r = n*T + t
  const float* ip = in + r * 3;
  x[i] = w[c * 3 + 0] * ip[0] + w[c * 3 + 1] * ip[1] + w[c * 3 + 2] * ip[2] + b[c];
}

// Edge scatter: nxt[dst, j] += cur[src, j] * invdeg[dst], 4 floats per thread
__global__ __launch_bounds__(256) void k_propagate4(const float* __restrict__ cur,
                                                    float* __restrict__ nxt,
                                                    const int* __restrict__ src,
                                                    const int* __restrict__ dst,
                                                    const float* __restrict__ invdeg,
                                                    int t8, long long total4) {
  long long i = (long long)blockIdx.x * blockDim.x + threadIdx.x;
  if (i >= total4) return;
  int e = (int)(i / t8);
  int q = (int)(i - (long long)e * t8);
  int s = src[e], dd = dst[e];
  float w = invdeg[dd];
  const float4 v = *(const float4*)(cur + ((size_t)s * t8 + q) * 4);
  float* p = nxt + ((size_t)dd * t8 + q) * 4;
  atomicAdd(p + 0, v.x * w);
  atomicAdd(p + 1, v.y * w);
  atomicAdd(p + 2, v.z * w);
  atomicAdd(p + 3, v.w * w);
}

// x_out[n,tt,c] += x_in[n, (Tin-tout)+tt, c]; float4 over channels (32%4==0)
__global__ __launch_bounds__(256) void k_residual4(float* xo,
                                                   const float* __restrict__ xi,
                                                   int Tin, int tout, long long total4) {
  long long i = (long long)blockIdx.x * blockDim.x + threadIdx.x;
  if (i >= total4) return;
  int c4 = (int)(i & 7);                      // 8 float4 per 32-channel row
  long long r = i >> 3;                       // (n,tt)
  long long n = r / tout;
  int tt = (int)(r - n * tout);
  float4* po = (float4*)xo + i;
  const float4* pi = (const float4*)xi + (n * Tin + (Tin - tout) + tt) * 8 + c4;
  float4 a = *po, b = *pi;
  a.x += b.x; a.y += b.y; a.z += b.z; a.w += b.w;
  *po = a;
}

// Per-channel sum / sumsq over [M,32] (stride multiple of 32 => fixed channel/thread)
__global__ __launch_bounds__(256) void k_bnstats(const float* __restrict__ x,
                                                 long long total, float* __restrict__ sums) {
  __shared__ float s1[256], s2[256];
  float a1 = 0.f, a2 = 0.f;
  long long stride = (long long)gridDim.x * blockDim.x;
  for (long long i = (long long)blockIdx.x * blockDim.x + threadIdx.x; i < total; i += stride) {
    float v = x[i]; a1 += v; a2 += v * v;
  }
  s1[threadIdx.x] = a1; s2[threadIdx.x] = a2;
  __syncthreads();
  if (threadIdx.x < 32) {
    float r1 = s1[threadIdx.x], r2 = s2[threadIdx.x];
#pragma unroll
    for (int w = 1; w < 8; ++w) { r1 += s1[threadIdx.x + 32 * w]; r2 += s2[threadIdx.x + 32 * w]; }
    atomicAdd(&sums[threadIdx.x], r1);
    atomicAdd(&sums[32 + threadIdx.x], r2);
  }
}

__global__ __launch_bounds__(256) void k_bnapply4(float* x,
                                                  const float* __restrict__ sums,
                                                  const float* __restrict__ g,
                                                  const float* __restrict__ b,
                                                  float cntinv, long long total4) {
  long long i = (long long)blockIdx.x * blockDim.x + threadIdx.x;
  if (i >= total4) return;
  int c0 = (int)((i * 4) & 31);
  float4* p = (float4*)x + i;
  float4 v = *p;
  float vv[4] = {v.x, v.y, v.z, v.w};
#pragma unroll
  for (int j = 0; j < 4; ++j) {
    int c = c0 + j;
    float mu  = sums[c] * cntinv;
    float var = sums[32 + c] * cntinv - mu * mu;
    float sc  = g[c] * rsqrtf(var + 1e-5f);
    vv[j] = (vv[j] - mu) * sc + b[c];
  }
  *p = make_float4(vv[0], vv[1], vv[2], vv[3]);
}

__global__ __launch_bounds__(256) void k_relu(float* x, long long n) {
  long long i = (long long)blockIdx.x * blockDim.x + threadIdx.x;
  if (i < n) x[i] = fmaxf(x[i], 0.f);
}

// ---------------------------------------------------------------------------
extern "C" void kernel_launch(void* const* d_in, const int* in_sizes, int n_in,
                              void* d_out, int out_size, void* d_ws, size_t ws_size,
                              hipStream_t stream) {
  (void)in_sizes; (void)n_in; (void)out_size; (void)ws_size;
  const float* inp     = (const float*)d_in[0];
  const int*   esrc    = (const int*)d_in[1];
  const int*   edst    = (const int*)d_in[2];
  const float* enter_w = (const float*)d_in[3];
  const float* enter_b = (const float*)d_in[4];
  const float* filt_w  = (const float*)d_in[5];
  const float* filt_b  = (const float*)d_in[6];
  const float* gate_w  = (const float*)d_in[7];
  const float* gate_b  = (const float*)d_in[8];
  const float* gc_w    = (const float*)d_in[9];
  const float* gc_b    = (const float*)d_in[10];
  const float* skip_w  = (const float*)d_in[11];
  const float* skip_b  = (const float*)d_in[12];
  const float* bn_g    = (const float*)d_in[13];
  const float* bn_b    = (const float*)d_in[14];
  const float* out1_w  = (const float*)d_in[15];
  const float* out1_b  = (const float*)d_in[16];
  const float* out2_w  = (const float*)d_in[17];
  const float* out2_b  = (const float*)d_in[18];
  float* out = (float*)d_out;

  float* ws = (float*)d_ws;
  size_t o = 0;
  float* xA     = ws + o; o += (size_t)NN * 32 * 13;
  float* xB     = ws + o; o += (size_t)NN * 32 * 13;
  float* h      = ws + o; o += (size_t)NN * 32 * 12;
  float* tmpA   = ws + o; o += (size_t)NN * 32 * 12;
  float* tmpB   = ws + o; o += (size_t)NN * 32 * 12;
  float* skip   = ws + o; o += (size_t)NN * 256;
  float* invdeg = ws + o; o += (size_t)NN;
  float* sums   = ws + o; o += 64;
  float* y1 = tmpA;   // reused after all layers (tmpA+tmpB contiguous >= N*512)

  auto blk = [](long long n) { return (unsigned)((n + 255) / 256); };

  // inverse degree (recomputed every call; deterministic)
  k_zero4   <<<blk(NN / 4), 256, 0, stream>>>((float4*)invdeg, NN / 4);
  k_degcount<<<blk(EE), 256, 0, stream>>>(edst, invdeg, EE);
  k_deginv  <<<blk(NN), 256, 0, stream>>>(invdeg, NN);

  // enter 1x1 conv -> xA [N,13,32]
  k_enter<<<blk((long long)NN * TT0 * 32), 256, 0, stream>>>(
      inp, enter_w, enter_b, xA, (long long)NN * TT0 * 32);

  // skip accumulator [N,256]
  k_zero4<<<blk((long long)NN * 64), 256, 0, stream>>>((float4*)skip, (long long)NN * 64);

  int dils[LLAY] = {1, 2, 1, 2, 1, 2, 1, 2};
  float* xin = xA; float* xout = xB;
  int Tin = TT0;
  for (int l = 0; l < LLAY; ++l) {
    int d = dils[l];
    int tout = Tin - d;
    long long M = (long long)NN * tout;   // always multiple of 16 (N = 16*1250)
    int t32 = tout * 32;

    // gated dilated conv -> h [M,32]
    {
      long long waves = (M / 16) * 2;
      k_gatedconv<<<blk(waves * 32), 256, 0, stream>>>(
          xin, Tin, d,
          filt_w + (size_t)l * 32 * 32 * 2, filt_b + l * 32,
          gate_w + (size_t)l * 32 * 32 * 2, gate_b + l * 32,
          h, tout, (int)M);
    }
    // skip += skip_w[l] @ h[:, last, :] + skip_b[l]  (only last column survives truncation)
    {
      long long waves = (NN / 16) * (256 / 16);
      k_gemm_t<1, 0, 1, 32><<<blk(waves * 32), 256, 0, stream>>>(
          h, t32, (tout - 1) * 32,
          skip_w + (size_t)l * 256 * 32, skip_b + l * 256,
          skip, skip, 256, NN, 256, 0);
    }

    if (l < LLAY - 1) {   // last layer's diffusion/BN feeds nothing downstream
      long long gw = (M / 16) * 2;
      // k = 0: xout = h @ gc_w[l,0]
      k_gemm_t<0, 0, 1, 32><<<blk(gw * 32), 256, 0, stream>>>(
          h, 32, 0, gc_w + (size_t)(l * 4 + 0) * 1024,
          nullptr, nullptr, xout, 32, (int)M, 32, 0);
      // k = 1..3: cur = propagate(cur); xout += cur @ gc_w[l,k]  (+gc_b at k=3)
      const float* cur = h; float* nx = tmpA;
      for (int k = 1; k < 4; ++k) {
        k_zero4<<<blk(M * 8), 256, 0, stream>>>((float4*)nx, M * 8);
        long long tot4 = (long long)EE * (t32 / 4);
        k_propagate4<<<blk(tot4), 256, 0, stream>>>(cur, nx, esrc, edst, invdeg, t32 / 4, tot4);
        const float* bias = (k == 3) ? (gc_b + l * 32) : nullptr;
        k_gemm_t<0, 0, 1, 32><<<blk(gw * 32), 256, 0, stream>>>(
            nx, 32, 0, gc_w + (size_t)(l * 4 + k) * 1024,
            bias, xout, xout, 32, (int)M, 32, 0);
        cur = nx;
        nx = (nx == tmpA) ? tmpB : tmpA;
      }
      // residual add + batchnorm (training-mode batch stats over N*t per channel)
      k_residual4<<<blk(M * 8), 256, 0, stream>>>(xout, xin, Tin, tout, M * 8);
      k_zero4   <<<1, 16, 0, stream>>>((float4*)sums, 16);
      k_bnstats <<<1024, 256, 0, stream>>>(xout, M * 32, sums);
      k_bnapply4<<<blk(M * 8), 256, 0, stream>>>(
          xout, sums, bn_g + l * 32, bn_b + l * 32, 1.0f / (float)M, M * 8);
      float* t = xin; xin = xout; xout = t;
    }
    Tin = tout;
  }

  // head: relu(skip) -> 512 (relu) -> 24
  k_relu<<<blk((long long)NN * 256), 256, 0, stream>>>(skip, (long long)NN * 256);
  {
    long long waves = (NN / 16) * (512 / 16);
    k_gemm_t<1, 0, 0, 256><<<blk(waves * 32), 256, 0, stream>>>(
        skip, 256, 0, out1_w, out1_b, nullptr, y1, 512, NN, 512, 1);
  }
  {
    long long waves = (NN / 16) * 2;   // ceil(24/16) = 2, masked tail
    k_gemm_t<1, 1, 0, 512><<<blk(waves * 32), 256, 0, stream>>>(
        y1, 512, 0, out2_w, out2_b, nullptr, out, 24, NN, 24, 0);
  }
}